// TextRecognitionHeadAttention_3719441679196
// MI455X (gfx1250) — compile-verified
//
#include <hip/hip_runtime.h>
#include <hip/hip_bf16.h>
#include <math.h>

typedef __attribute__((ext_vector_type(2))) float v2f;
typedef __attribute__((ext_vector_type(8))) float v8f;

#define HID 512
#define VOCAB 256
#define SEQLEN 28
#define TPOS 256   /* 8*32 spatial positions */
#define BATCH 64

#if __has_builtin(__builtin_amdgcn_global_load_async_to_lds_b32)
#define HAVE_ASYNC_LDS 1
typedef __attribute__((address_space(1))) int* as1_i32p;
typedef __attribute__((address_space(3))) int* as3_i32p;
#endif

// ---------------------------------------------------------------------------
// fp32 WMMA 16x16x4 helper. A: 16x4 (lane l holds row m=l%16, K pair at
// koff = (l>=16)?2:0). B: 4x16 (lane l holds col n=l%16, same K pair).
// C/D: v8f, lane l vgpr i -> row (i + (l>=16?8:0)), col l%16.
// ---------------------------------------------------------------------------
__device__ __forceinline__ v8f wmma4(v2f a, v2f b, v8f c) {
  return __builtin_amdgcn_wmma_f32_16x16x4_f32(false, a, false, b,
                                               (short)0, c, false, false);
}

// A row-major [M x K] via per-lane row pointer pa, B given as weight rows
// (row n contiguous over K) via per-lane row pointer pw. Both K-contiguous,
// 8-byte aligned -> float2 loads. Loads batched 4-deep ahead of 4 WMMAs so a
// single wait covers a burst of 8 vmem loads.
__device__ __forceinline__ v8f gemm_rows(const float* __restrict__ pa,
                                         const float* __restrict__ pw,
                                         int K, int koff, v8f acc) {
#pragma unroll 2
  for (int k0 = 0; k0 < K; k0 += 16) {
    v2f av[4], bv[4];
#pragma unroll
    for (int u = 0; u < 4; ++u) {
      int k = k0 + 4 * u + koff;
      av[u] = *(const v2f*)(pa + k);
      bv[u] = *(const v2f*)(pw + k);
    }
#pragma unroll
    for (int u = 0; u < 4; ++u) acc = wmma4(av[u], bv[u], acc);
  }
  return acc;
}

__device__ __forceinline__ float sigmoidf_(float x) {
  return 1.0f / (1.0f + expf(-x));
}

// ---------------------------------------------------------------------------
// Encoder conv3x3 SAME + bias + BN(eval) + ReLU, 512->512, implicit GEMM.
// M = 64*8*32 = 16384 pixels, N = 512, K = 512*9 (tap-major: 9 taps x 512 ci).
// One wave per 16x16 D tile; 8 waves/WG share an n-tile; weights for the
// current tap staged in LDS (512x16 f32 = 32KB) via async global->LDS when
// available. SAME padding is branchless: clamped address + 0/1 mask multiply.
// Grid: 4096 blocks x 256 threads.  write_enc: 0 -> NCHW out, 1 -> [B,T,H].
// ---------------------------------------------------------------------------
__global__ __launch_bounds__(256) void conv_bn_relu_kernel(
    const float* __restrict__ X,    // [64,512,8,32]
    const float* __restrict__ Wc,   // [512,512,3,3] (layer slice)
    const float* __restrict__ cb, const float* __restrict__ gam,
    const float* __restrict__ bet, const float* __restrict__ mu,
    const float* __restrict__ var,
    float* __restrict__ Y, int write_enc) {
  __shared__ float Bs[512 * 16];

  const int lane = threadIdx.x & 31;
  const int wave = threadIdx.x >> 5;
  const int col  = lane & 15;
  const int koff = (lane >> 4) << 1;          // 0 or 2
  const int ntile = blockIdx.x & 31;          // 32 n-tiles of 16 channels
  const int mtile = (blockIdx.x >> 5) * 8 + wave;  // 0..1023 pixel tiles
  const int x0 = (mtile & 1) * 16;
  const int yy = (mtile >> 1) & 7;
  const int bb = mtile >> 4;
  const int xm = x0 + col;                    // A-row pixel x (m = col)

  v8f acc = {};
  for (int tap = 0; tap < 9; ++tap) {
    const int dy = tap / 3 - 1, dx = tap % 3 - 1;
    __syncthreads();
    // stage B tile for this tap: Bs[ci*16 + n] = Wc[(nglob*512+ci)*9 + tap]
#pragma unroll
    for (int r = 0; r < 32; ++r) {
      int idx = threadIdx.x + r * 256;        // 0..8191
      int n = idx >> 9;                       // 0..15
      int ci = idx & 511;
      const float* gp = &Wc[(((ntile * 16 + n) * 512 + ci) * 9) + tap];
#ifdef HAVE_ASYNC_LDS
      __builtin_amdgcn_global_load_async_to_lds_b32(
          (as1_i32p)gp, (as3_i32p)&Bs[ci * 16 + n], 0, 0);
#else
      Bs[ci * 16 + n] = *gp;
#endif
    }
#ifdef HAVE_ASYNC_LDS
#if __has_builtin(__builtin_amdgcn_s_wait_asynccnt)
    __builtin_amdgcn_s_wait_asynccnt(0);
#else
    asm volatile("s_wait_asynccnt 0x0" ::: "memory");
#endif
#endif
    __syncthreads();

    // branchless SAME padding: clamp address into bounds, multiply by mask
    const int ya = yy + dy;
    const int xa = xm + dx;
    const bool ok = (ya >= 0) && (ya < 8) && (xa >= 0) && (xa < 32);
    const float msk = ok ? 1.0f : 0.0f;
    const int yc = min(max(ya, 0), 7);
    const int xc = min(max(xa, 0), 31);
    const long abase = (long)bb * 131072 + (long)yc * 32 + xc;  // ci stride 256
#pragma unroll 2
    for (int k0 = 0; k0 < 512; k0 += 16) {
      v2f av[4], bv[4];
#pragma unroll
      for (int u = 0; u < 4; ++u) {
        int ci = k0 + 4 * u + koff;
        av[u].x = X[abase + (long)ci * 256] * msk;
        av[u].y = X[abase + (long)(ci + 1) * 256] * msk;
        bv[u].x = Bs[ci * 16 + col];
        bv[u].y = Bs[(ci + 1) * 16 + col];
      }
#pragma unroll
      for (int u = 0; u < 4; ++u) acc = wmma4(av[u], bv[u], acc);
    }
  }

  // epilogue: +conv_b, BN(eval), ReLU
  const int n = ntile * 16 + col;
  const float sc = gam[n] * rsqrtf(var[n] + 1e-5f);
  const float sh = cb[n] - mu[n];
  const float bt = bet[n];
  const int rbase = (lane >> 4) << 3;         // +8 for upper half-wave
#pragma unroll
  for (int i = 0; i < 8; ++i) {
    int row = i + rbase;                      // pixel x offset in tile
    float vv = sc * (acc[i] + sh) + bt;
    vv = fmaxf(vv, 0.0f);
    int x = x0 + row;
    if (write_enc) {
      // enc[b][t][h], t = y*32+x
      Y[((bb * 256) + yy * 32 + x) * 512 + n] = vv;
    } else {
      // NCHW
      Y[(((bb * 512) + n) * 8 + yy) * 32 + x] = vv;
    }
  }
}

// ---------------------------------------------------------------------------
// ew = enc @ Wenc^T + benc : M=16384 (b*256+t), N=512, K=512.
// Grid: 4096 blocks x 256 threads (32768 wave-tiles).
// ---------------------------------------------------------------------------
__global__ __launch_bounds__(256) void ew_gemm_kernel(
    const float* __restrict__ enc, const float* __restrict__ Wenc,
    const float* __restrict__ benc, float* __restrict__ ew) {
  const int lane = threadIdx.x & 31;
  const int wave = threadIdx.x >> 5;
  const int col  = lane & 15;
  const int koff = (lane >> 4) << 1;
  const int id = blockIdx.x * 8 + wave;       // 0..32767
  const int ntile = id & 31;
  const int mtile = id >> 5;                  // 0..1023
  const float* pa = enc + (mtile * 16 + col) * 512;
  const float* pw = Wenc + (ntile * 16 + col) * 512;
  v8f acc = {};
  acc = gemm_rows(pa, pw, 512, koff, acc);
  const int nj = ntile * 16 + col;
  const float bn = benc[nj];
  const int rbase = (lane >> 4) << 3;
#pragma unroll
  for (int i = 0; i < 8; ++i) {
    int row = mtile * 16 + i + rbase;
    ew[row * 512 + nj] = acc[i] + bn;
  }
}

// ---------------------------------------------------------------------------
// init: h0 = 0, inp = SOS(=1)
// ---------------------------------------------------------------------------
__global__ void init_kernel(float* __restrict__ h0, int* __restrict__ inp) {
  int i = blockIdx.x * 256 + threadIdx.x;
  if (i < BATCH * HID) h0[i] = 0.0f;
  if (i < BATCH) inp[i] = 1;
}

// ---------------------------------------------------------------------------
// Per-step attention + combine: one WG per batch element (64 x 256 threads).
//  hw = Whid@h + bhid ; s_t = v . tanh(ew + hw) ; softmax ; ctx = a^T enc ;
//  o = relu(Wcomb @ [x; ctx] + bcomb)
// ---------------------------------------------------------------------------
__global__ __launch_bounds__(256) void attn_kernel(
    const float* __restrict__ ew, const float* __restrict__ enc,
    const float* __restrict__ h, const float* __restrict__ Whid,
    const float* __restrict__ bhid, const float* __restrict__ v,
    const float* __restrict__ emb, const int* __restrict__ inp,
    const float* __restrict__ Wcomb, const float* __restrict__ bcomb,
    float* __restrict__ o_out) {
  __shared__ float sh_h[512];
  __shared__ float sh_hw[512];
  __shared__ float sh_a[256];
  __shared__ float sh_x[512];
  __shared__ float sh_cx[512];
  __shared__ float sred[16];

  const int b = blockIdx.x;
  const int tid = threadIdx.x;

  sh_h[tid] = h[b * 512 + tid];
  sh_h[tid + 256] = h[b * 512 + 256 + tid];
  const int tok = inp[b];
  sh_x[tid] = emb[tok * 512 + tid];
  sh_x[tid + 256] = emb[tok * 512 + 256 + tid];
  __syncthreads();

  // hw = Whid @ h + bhid
#pragma unroll
  for (int jj = 0; jj < 2; ++jj) {
    int j = tid + jj * 256;
    const float* wr = Whid + j * 512;
    float s = bhid[j];
#pragma unroll 8
    for (int k = 0; k < 512; ++k) s += wr[k] * sh_h[k];
    sh_hw[j] = s;
  }
  __syncthreads();

  // score for t = tid
  {
    const float* ewp = ew + ((b * 256 + tid) * 512);
    float s = 0.0f;
#pragma unroll 4
    for (int k = 0; k < 512; ++k) s += v[k] * tanhf(ewp[k] + sh_hw[k]);
    sh_a[tid] = s;
  }
  __syncthreads();

  // softmax over 256
  float mv = sh_a[tid];
  for (int off = 16; off; off >>= 1) mv = fmaxf(mv, __shfl_down(mv, off, 32));
  if ((tid & 31) == 0) sred[tid >> 5] = mv;
  __syncthreads();
  if (tid == 0) {
    float mm = sred[0];
    for (int i = 1; i < 8; ++i) mm = fmaxf(mm, sred[i]);
    sred[8] = mm;
  }
  __syncthreads();
  const float M = sred[8];
  float e = expf(sh_a[tid] - M);
  float sv = e;
  for (int off = 16; off; off >>= 1) sv += __shfl_down(sv, off, 32);
  if ((tid & 31) == 0) sred[tid >> 5] = sv;
  __syncthreads();
  if (tid == 0) {
    float s = sred[0];
    for (int i = 1; i < 8; ++i) s += sred[i];
    sred[9] = s;
  }
  __syncthreads();
  const float S = sred[9];
  sh_a[tid] = e / S;
  __syncthreads();

  // ctx
#pragma unroll
  for (int jj = 0; jj < 2; ++jj) {
    int j = tid + jj * 256;
    const float* ep = enc + b * 256 * 512 + j;
    float c = 0.0f;
#pragma unroll 4
    for (int t = 0; t < 256; ++t) c += sh_a[t] * ep[t * 512];
    sh_cx[j] = c;
  }
  __syncthreads();

  // o = relu(Wcomb @ [x; ctx] + bcomb)
#pragma unroll
  for (int jj = 0; jj < 2; ++jj) {
    int j = tid + jj * 256;
    const float* wr = Wcomb + j * 1024;
    float s = bcomb[j];
#pragma unroll 8
    for (int k = 0; k < 512; ++k) s += wr[k] * sh_x[k];
#pragma unroll 8
    for (int k = 0; k < 512; ++k) s += wr[512 + k] * sh_cx[k];
    o_out[b * 512 + j] = fmaxf(s, 0.0f);
  }
}

// ---------------------------------------------------------------------------
// GRU step (WMMA): gates r,z,n from o@Wih^T and h@Whh^T, writes h_next.
// 128 wave-tiles (M=64 -> 4, N=512 -> 32). Grid: 16 x 256.
// ---------------------------------------------------------------------------
__global__ __launch_bounds__(256) void gru_kernel(
    const float* __restrict__ o, const float* __restrict__ h,
    const float* __restrict__ Wih, const float* __restrict__ Whh,
    const float* __restrict__ bih, const float* __restrict__ bhh,
    float* __restrict__ hn) {
  const int lane = threadIdx.x & 31;
  const int wave = threadIdx.x >> 5;
  const int col  = lane & 15;
  const int koff = (lane >> 4) << 1;
  const int id = blockIdx.x * 8 + wave;       // 0..127
  const int ntile = id & 31;
  const int mtile = id >> 5;                  // 0..3
  const int mb = mtile * 16, nb = ntile * 16;
  const int nj = nb + col;

  const float* paO = o + (mb + col) * 512;
  const float* paH = h + (mb + col) * 512;

  v8f aR = {}, aZ = {}, aNi = {}, aNh = {};
  aR = gemm_rows(paO, Wih + nj * 512, 512, koff, aR);
  aR = gemm_rows(paH, Whh + nj * 512, 512, koff, aR);
  aZ = gemm_rows(paO, Wih + (512 + nj) * 512, 512, koff, aZ);
  aZ = gemm_rows(paH, Whh + (512 + nj) * 512, 512, koff, aZ);
  aNi = gemm_rows(paO, Wih + (1024 + nj) * 512, 512, koff, aNi);
  aNh = gemm_rows(paH, Whh + (1024 + nj) * 512, 512, koff, aNh);

  const float bR = bih[nj] + bhh[nj];
  const float bZ = bih[512 + nj] + bhh[512 + nj];
  const float bNi = bih[1024 + nj];
  const float bNh = bhh[1024 + nj];
  const int rbase = (lane >> 4) << 3;
#pragma unroll
  for (int i = 0; i < 8; ++i) {
    int brow = mb + i + rbase;
    float r = sigmoidf_(aR[i] + bR);
    float z = sigmoidf_(aZ[i] + bZ);
    float nn = tanhf(aNi[i] + bNi + r * (aNh[i] + bNh));
    float hv = h[brow * 512 + nj];
    hn[brow * 512 + nj] = (1.0f - z) * nn + z * hv;
  }
}

// ---------------------------------------------------------------------------
// out = h2 @ Wout^T + bout : M=64, N=256, K=512 -> 64 wave-tiles, 8 x 256.
// ---------------------------------------------------------------------------
__global__ __launch_bounds__(256) void out_gemm_kernel(
    const float* __restrict__ h2, const float* __restrict__ Wout,
    const float* __restrict__ bout, float* __restrict__ out) {
  const int lane = threadIdx.x & 31;
  const int wave = threadIdx.x >> 5;
  const int col  = lane & 15;
  const int koff = (lane >> 4) << 1;
  const int id = blockIdx.x * 8 + wave;       // 0..63
  const int ntile = id & 15;
  const int mtile = id >> 4;                  // 0..3
  const float* pa = h2 + (mtile * 16 + col) * 512;
  const float* pw = Wout + (ntile * 16 + col) * 512;
  v8f acc = {};
  acc = gemm_rows(pa, pw, 512, koff, acc);
  const int nj = ntile * 16 + col;
  const float bo = bout[nj];
  const int rbase = (lane >> 4) << 3;
#pragma unroll
  for (int i = 0; i < 8; ++i) {
    int brow = mtile * 16 + i + rbase;
    out[brow * 256 + nj] = acc[i] + bo;
  }
}

// ---------------------------------------------------------------------------
// argmax over 256 vocab per batch element (first-index tie-break).
// ---------------------------------------------------------------------------
__global__ void argmax_kernel(const float* __restrict__ out,
                              int* __restrict__ inp) {
  const int b = blockIdx.x;
  const int lane = threadIdx.x;
  const float* p = out + b * 256;
  float bv = -3.4e38f;
  int bi = 0;
#pragma unroll
  for (int r = 0; r < 8; ++r) {
    int j = lane * 8 + r;
    float vv = p[j];
    if (vv > bv) { bv = vv; bi = j; }
  }
  for (int off = 16; off; off >>= 1) {
    float ov = __shfl_down(bv, off, 32);
    int oi = __shfl_down(bi, off, 32);
    if (ov > bv || (ov == bv && oi < bi)) { bv = ov; bi = oi; }
  }
  if (lane == 0) inp[b] = bi;
}

// ---------------------------------------------------------------------------
extern "C" void kernel_launch(void* const* d_in, const int* in_sizes, int n_in,
                              void* d_out, int out_size, void* d_ws,
                              size_t ws_size, hipStream_t stream) {
  const float* features = (const float*)d_in[0];
  const float* conv_w   = (const float*)d_in[1];   // [2,512,512,3,3]
  const float* conv_b   = (const float*)d_in[2];   // [2,512]
  const float* bn_gamma = (const float*)d_in[3];
  const float* bn_beta  = (const float*)d_in[4];
  const float* bn_mean  = (const float*)d_in[5];
  const float* bn_var   = (const float*)d_in[6];
  const float* emb      = (const float*)d_in[7];   // [256,512]
  const float* Wih      = (const float*)d_in[8];   // [1536,512]
  const float* Whh      = (const float*)d_in[9];   // [1536,512]
  const float* bih      = (const float*)d_in[10];
  const float* bhh      = (const float*)d_in[11];
  const float* Wenc     = (const float*)d_in[12];  // [512,512]
  const float* benc     = (const float*)d_in[13];
  const float* Whid     = (const float*)d_in[14];  // [512,512]
  const float* bhid     = (const float*)d_in[15];
  const float* vvec     = (const float*)d_in[16];  // [512,1]
  const float* Wcomb    = (const float*)d_in[17];  // [512,1024]
  const float* bcomb    = (const float*)d_in[18];
  const float* Wout     = (const float*)d_in[19];  // [256,512]
  const float* bout     = (const float*)d_in[20];
  float* outp = (float*)d_out;                     // [28,64,256]

  // workspace layout
  char* ws = (char*)d_ws;
  const size_t SZ_FEAT = (size_t)BATCH * HID * TPOS * sizeof(float);  // 33.5MB
  float* f1  = (float*)(ws);                       // conv layer-0 out (NCHW)
  float* enc = (float*)(ws + SZ_FEAT);             // [B,T,H]
  float* ew  = (float*)(ws + 2 * SZ_FEAT);         // [B,T,H]
  float* h0  = (float*)(ws + 3 * SZ_FEAT);
  float* h1  = (float*)(ws + 3 * SZ_FEAT + (size_t)BATCH * HID * sizeof(float));
  float* obuf= (float*)(ws + 3 * SZ_FEAT + 2 * (size_t)BATCH * HID * sizeof(float));
  int*   inp = (int*)  (ws + 3 * SZ_FEAT + 3 * (size_t)BATCH * HID * sizeof(float));

  const size_t LAYER_W = (size_t)HID * HID * 9;

  // encoder layer 0: features -> f1 (NCHW)
  conv_bn_relu_kernel<<<4096, 256, 0, stream>>>(
      features, conv_w, conv_b, bn_gamma, bn_beta, bn_mean, bn_var, f1, 0);
  // encoder layer 1: f1 -> enc ([B,T,H])
  conv_bn_relu_kernel<<<4096, 256, 0, stream>>>(
      f1, conv_w + LAYER_W, conv_b + HID, bn_gamma + HID, bn_beta + HID,
      bn_mean + HID, bn_var + HID, enc, 1);
  // step-invariant ew = enc @ Wenc^T + benc
  ew_gemm_kernel<<<4096, 256, 0, stream>>>(enc, Wenc, benc, ew);
  // h0 = 0, inp = SOS
  init_kernel<<<128, 256, 0, stream>>>(h0, inp);

  for (int step = 0; step < SEQLEN; ++step) {
    float* hc = (step & 1) ? h1 : h0;
    float* hn = (step & 1) ? h0 : h1;
    attn_kernel<<<BATCH, 256, 0, stream>>>(ew, enc, hc, Whid, bhid, vvec, emb,
                                           inp, Wcomb, bcomb, obuf);
    gru_kernel<<<16, 256, 0, stream>>>(obuf, hc, Wih, Whh, bih, bhh, hn);
    out_gemm_kernel<<<8, 256, 0, stream>>>(hn, Wout, bout,
                                           outp + (size_t)step * BATCH * VOCAB);
    argmax_kernel<<<BATCH, 32, 0, stream>>>(outp + (size_t)step * BATCH * VOCAB,
                                            inp);
  }
}